// HRVQ_15771119911012
// MI455X (gfx1250) — compile-verified
//
#include <hip/hip_runtime.h>
#include <float.h>

typedef __attribute__((ext_vector_type(2))) float v2f;
typedef __attribute__((ext_vector_type(4))) float v4f;
typedef __attribute__((ext_vector_type(8))) float v8f;

#define NL    3
#define KC    512
#define DIM   1024
#define NTOK  65536          // 16 * 4096
#define ROWS  16
#define RSTRIDE 1028         // 1024 + 4 pad -> conflict-free A-operand LDS reads
#define DECAYF 0.99f
#define OMDF   0.01f
#define EPSF   1e-5f

// ---------------------------------------------------------------------------
// Precompute |C_k|^2 once per code row (NL*KC rows). One wave per row.
// ---------------------------------------------------------------------------
__global__ __launch_bounds__(32) void hrvq_code_norms(
    const float* __restrict__ codebooks,   // [NL*KC, DIM]
    float* __restrict__ cnorms)            // [NL*KC]
{
    const int row  = blockIdx.x;
    const int lane = threadIdx.x;
    const float* c = codebooks + (size_t)row * DIM;
    float s = 0.f;
    #pragma unroll 4
    for (int k = lane * 4; k < DIM; k += 128) {
        const v4f v = *(const v4f*)(c + k);
        s = fmaf(v.x, v.x, s); s = fmaf(v.y, v.y, s);
        s = fmaf(v.z, v.z, s); s = fmaf(v.w, v.w, s);
    }
    #pragma unroll
    for (int m = 1; m < 32; m <<= 1) s += __shfl_xor(s, m, 32);
    if (lane == 0) cnorms[row] = s;
}

// ---------------------------------------------------------------------------
// Fused 3-level residual VQ: per block = 16 tokens, 8 waves x 64 codes each.
// WMMA f32 16x16x4 computes dot(residual_row, code_row); distance = |C|^2-2dot.
// Hot loop is pure WMMA + ds/global loads (norms precomputed).
// ---------------------------------------------------------------------------
__global__ __launch_bounds__(256) void hrvq_main(
    const float* __restrict__ z_e,        // [NTOK, DIM]
    const float* __restrict__ codebooks,  // [NL, KC, DIM]
    const float* __restrict__ cnorms,     // [NL, KC]
    float* __restrict__ zq_out,           // [NTOK, DIM]
    float* __restrict__ idx_out,          // [NL, NTOK] (as float)
    float* __restrict__ counts,           // [NL, KC]   (zeroed)
    float* __restrict__ sums,             // [NL, KC, DIM] (zeroed)
    float* __restrict__ loss_acc)         // [1] (zeroed)
{
    __shared__ float rs[ROWS * RSTRIDE];  // residual tile, updated per level
    __shared__ float wbd[8][16];          // per-wave best distance per row
    __shared__ int   wbi[8][16];          // per-wave best code per row
    __shared__ int   fidx[16];            // final index per row
    __shared__ float lred[8];             // loss partials per wave

    const int tid  = threadIdx.x;
    const int wave = tid >> 5;
    const int lane = tid & 31;
    const int mn   = lane & 15;           // A row (m) and B col (n) index
    const int half = lane >> 4;
    const int row_base = blockIdx.x * ROWS;

    // ---- stage z_e tile into LDS (initial residual), non-temporal stream ----
    {
        const int r   = tid >> 4;
        const int c16 = tid & 15;
        const float* src = z_e + (size_t)(row_base + r) * DIM;
        for (int j = 0; j < 16; ++j) {
            const int dim = c16 * 4 + j * 64;
            const v4f v = __builtin_nontemporal_load((const v4f*)(src + dim));
            *(v4f*)(&rs[r * RSTRIDE + dim]) = v;
        }
    }

    v4f zq[16];
    #pragma unroll
    for (int j = 0; j < 16; ++j) zq[j] = (v4f){0.f, 0.f, 0.f, 0.f};
    float lsum = 0.f;
    const float commit[NL] = {0.25f, 0.5f, 1.0f};

    __syncthreads();

    for (int l = 0; l < NL; ++l) {
        const float* Cl = codebooks + (size_t)l * KC * DIM;

        float bestd[8];
        int   besti[8];
        #pragma unroll
        for (int v = 0; v < 8; ++v) { bestd[v] = FLT_MAX; besti[v] = 0; }

        // ---- distances via WMMA over this wave's 64 codes ----
        for (int ct = 0; ct < 4; ++ct) {
            const int code = wave * 64 + ct * 16 + mn;
            const float cn = cnorms[l * KC + code];       // precomputed |C_n|^2
            const float* crow = Cl + (size_t)code * DIM + 2 * half;
            const float* arow = &rs[mn * RSTRIDE + 2 * half];

            v8f acc = {0.f, 0.f, 0.f, 0.f, 0.f, 0.f, 0.f, 0.f};
            #pragma unroll 8
            for (int k = 0; k < DIM; k += 4) {
                const v2f a = *(const v2f*)(arow + k);  // A 16x4: half0 K0,K1; half1 K2,K3
                const v2f b = *(const v2f*)(crow + k);  // B 4x16: same striping, n = lane%16
                acc = __builtin_amdgcn_wmma_f32_16x16x4_f32(
                        false, a, false, b, (short)0, acc, false, false);
            }

            #pragma unroll
            for (int v = 0; v < 8; ++v) {          // D[m][n]: m = half*8+v, n = mn
                const float d = cn - 2.f * acc[v]; // row norm constant: argmin-invariant
                if (d < bestd[v] || (d == bestd[v] && code < besti[v])) {
                    bestd[v] = d; besti[v] = code;
                }
            }
        }

        // ---- argmin across the 16 lanes of each half (butterfly) ----
        #pragma unroll
        for (int mask = 1; mask <= 8; mask <<= 1) {
            #pragma unroll
            for (int v = 0; v < 8; ++v) {
                const float od = __shfl_xor(bestd[v], mask, 32);
                const int   oi = __shfl_xor(besti[v], mask, 32);
                if (od < bestd[v] || (od == bestd[v] && oi < besti[v])) {
                    bestd[v] = od; besti[v] = oi;
                }
            }
        }
        if (mn == 0) {                              // lanes 0 and 16
            #pragma unroll
            for (int v = 0; v < 8; ++v) {
                wbd[wave][half * 8 + v] = bestd[v];
                wbi[wave][half * 8 + v] = besti[v];
            }
        }
        __syncthreads();

        // ---- cross-wave argmin, emit index + count ----
        if (tid < 16) {
            float bd = wbd[0][tid]; int bi = wbi[0][tid];
            for (int w = 1; w < 8; ++w) {
                const float od = wbd[w][tid]; const int oi = wbi[w][tid];
                if (od < bd || (od == bd && oi < bi)) { bd = od; bi = oi; }
            }
            fidx[tid] = bi;
            idx_out[(size_t)l * NTOK + row_base + tid] = (float)bi;
            atomicAdd(&counts[l * KC + bi], 1.0f);
        }
        __syncthreads();

        // ---- update: loss, scatter sums(pre-update residual), residual-=q, zq+=q
        {
            const int r   = tid >> 4;
            const int c16 = tid & 15;
            const int fi  = fidx[r];
            const float* qrow = Cl + (size_t)fi * DIM;
            float* srow = sums + ((size_t)l * KC + fi) * DIM;
            const float cl = commit[l];
            #pragma unroll 4
            for (int j = 0; j < 16; ++j) {
                const int dim = c16 * 4 + j * 64;
                const v4f rv = *(const v4f*)(&rs[r * RSTRIDE + dim]);
                const v4f qv = *(const v4f*)(qrow + dim);
                const v4f dv = rv - qv;
                lsum += cl * (dv.x * dv.x + dv.y * dv.y + dv.z * dv.z + dv.w * dv.w);
                atomicAdd(srow + dim + 0, rv.x);
                atomicAdd(srow + dim + 1, rv.y);
                atomicAdd(srow + dim + 2, rv.z);
                atomicAdd(srow + dim + 3, rv.w);
                zq[j] += qv;
                *(v4f*)(&rs[r * RSTRIDE + dim]) = dv;
            }
        }
        __syncthreads();
    }

    // ---- write z_q (== z_q_st forward value), non-temporal stream ----
    {
        const int r   = tid >> 4;
        const int c16 = tid & 15;
        float* dst = zq_out + (size_t)(row_base + r) * DIM;
        #pragma unroll 4
        for (int j = 0; j < 16; ++j) {
            const int dim = c16 * 4 + j * 64;
            __builtin_nontemporal_store(zq[j], (v4f*)(dst + dim));
        }
    }

    // ---- block loss reduction ----
    #pragma unroll
    for (int mask = 1; mask < 32; mask <<= 1) lsum += __shfl_xor(lsum, mask, 32);
    if (lane == 0) lred[wave] = lsum;
    __syncthreads();
    if (tid == 0) {
        float t = 0.f;
        for (int w = 0; w < 8; ++w) t += lred[w];
        atomicAdd(loss_acc, t * (1.0f / ((float)NTOK * (float)DIM)));
    }
}

// ---------------------------------------------------------------------------
// EMA cluster stats: cs = decay*ema + (1-decay)*counts; n = sum_k cs
// ---------------------------------------------------------------------------
__global__ __launch_bounds__(KC) void hrvq_ema_stats(
    const float* __restrict__ ema_cluster,  // [NL, KC]
    const float* __restrict__ counts,       // [NL, KC]
    float* __restrict__ cs_out,             // [NL, KC]
    float* __restrict__ n_out)              // [NL]
{
    __shared__ float red[KC];
    const int l = blockIdx.x;
    const int k = threadIdx.x;
    const float cs = DECAYF * ema_cluster[l * KC + k] + OMDF * counts[l * KC + k];
    cs_out[l * KC + k] = cs;
    red[k] = cs;
    __syncthreads();
    for (int s = KC / 2; s > 0; s >>= 1) {
        if (k < s) red[k] += red[k + s];
        __syncthreads();
    }
    if (k == 0) n_out[l] = red[0];
}

// ---------------------------------------------------------------------------
// new_codebook = (decay*ema_w + (1-decay)*sums) / laplace(cs, n)
// ---------------------------------------------------------------------------
__global__ __launch_bounds__(256) void hrvq_codebook_out(
    const float* __restrict__ ema_w,   // [NL, KC, DIM]
    const float* __restrict__ sums,    // [NL, KC, DIM]
    const float* __restrict__ cs,      // [NL, KC]
    const float* __restrict__ nval,    // [NL]
    float* __restrict__ cb_out)        // [NL, KC, DIM]
{
    const size_t i = (size_t)blockIdx.x * blockDim.x + threadIdx.x;
    const int l = (int)(i / ((size_t)KC * DIM));
    const int k = (int)((i / DIM) % KC);
    const float n = nval[l];
    const float c = cs[l * KC + k];
    const float cs_norm = (c + EPSF) / (n + (float)KC * EPSF) * n;
    const float w = DECAYF * ema_w[i] + OMDF * sums[i];
    cb_out[i] = w / cs_norm;
}

__global__ void hrvq_loss_copy(const float* __restrict__ loss_acc,
                               float* __restrict__ out)
{
    out[0] = loss_acc[0];
}

// ---------------------------------------------------------------------------
extern "C" void kernel_launch(void* const* d_in, const int* in_sizes, int n_in,
                              void* d_out, int out_size, void* d_ws, size_t ws_size,
                              hipStream_t stream) {
    (void)in_sizes; (void)n_in; (void)out_size; (void)ws_size;

    const float* z_e         = (const float*)d_in[0];  // [16,4096,1024]
    const float* codebooks   = (const float*)d_in[1];  // [3,512,1024]
    const float* ema_cluster = (const float*)d_in[2];  // [3,512]
    const float* ema_w       = (const float*)d_in[3];  // [3,512,1024]

    float* out = (float*)d_out;
    float* zq_out   = out;                                   // NTOK*DIM
    float* loss_out = out + (size_t)NTOK * DIM;              // 1
    float* idx_out  = loss_out + 1;                          // NL*NTOK
    float* cb_out   = idx_out + (size_t)NL * NTOK;           // NL*KC*DIM

    float* ws       = (float*)d_ws;
    float* sums     = ws;                                    // NL*KC*DIM
    float* counts   = sums + (size_t)NL * KC * DIM;          // NL*KC
    float* loss_acc = counts + NL * KC;                      // 1
    float* cs       = loss_acc + 1;                          // NL*KC
    float* nval     = cs + NL * KC;                          // NL
    float* cnorms   = nval + NL;                             // NL*KC

    // zero the accumulated stats region (sums + counts + loss)
    hipMemsetAsync(ws, 0,
                   ((size_t)NL * KC * DIM + NL * KC + 1) * sizeof(float),
                   stream);

    hrvq_code_norms<<<NL * KC, 32, 0, stream>>>(codebooks, cnorms);

    hrvq_main<<<NTOK / ROWS, 256, 0, stream>>>(
        z_e, codebooks, cnorms, zq_out, idx_out, counts, sums, loss_acc);

    hrvq_ema_stats<<<NL, KC, 0, stream>>>(ema_cluster, counts, cs, nval);

    hrvq_codebook_out<<<(NL * KC * DIM) / 256, 256, 0, stream>>>(
        ema_w, sums, cs, nval, cb_out);

    hrvq_loss_copy<<<1, 1, 0, stream>>>(loss_acc, loss_out);
}